// _SGCModule_51213190037917
// MI455X (gfx1250) — compile-verified
//
#include <hip/hip_runtime.h>

#define NN 100000
#define NE 1600000
#define C  128

typedef __attribute__((ext_vector_type(2))) float v2f;
typedef __attribute__((ext_vector_type(8))) float v8f;

// Native CDNA5 f32 atomic add at device scope (RMW performed at L2).
// No-return form -> tracked with STOREcnt; s_endpgm wait-idles, and these are
// the last memory ops in their kernels, so compiler waitcnt bookkeeping stays valid.
__device__ __forceinline__ void atomAddF(float* p, float v) {
    asm volatile("global_atomic_add_f32 %0, %1, off scope:SCOPE_DEV"
                 :: "v"(p), "v"(v)
                 : "memory");
}

// deg[i] = 1.0 (self loop contribution)
__global__ void __launch_bounds__(256) deg_init_kernel(float* __restrict__ deg) {
    int i = blockIdx.x * 256 + threadIdx.x;
    if (i < NN) deg[i] = 1.0f;
}

// deg[dst[e]] += 1
__global__ void __launch_bounds__(256) deg_count_kernel(const long long* __restrict__ ei,
                                                        float* __restrict__ deg) {
    int e = blockIdx.x * 256 + threadIdx.x;
    if (e < NE) {
        int d = (int)ei[(size_t)NE + e];
        atomAddF(&deg[d], 1.0f);
    }
}

// in-place deg -> dinv
__global__ void __launch_bounds__(256) dinv_kernel(float* __restrict__ deg) {
    int i = blockIdx.x * 256 + threadIdx.x;
    if (i < NN) {
        float dg = deg[i];
        deg[i] = (dg > 0.0f) ? rsqrtf(dg) : 0.0f;
    }
}

// norm[e] = dinv[src] * dinv[dst]
__global__ void __launch_bounds__(256) norm_kernel(const long long* __restrict__ ei,
                                                   const float* __restrict__ dinv,
                                                   float* __restrict__ nrm) {
    int e = blockIdx.x * 256 + threadIdx.x;
    if (e < NE) {
        int s = (int)ei[e];
        int d = (int)ei[(size_t)NE + e];
        nrm[e] = dinv[s] * dinv[d];
    }
}

// hout[i] = dinv[i]^2 * hin[i]   (self-loop term; plain stores initialize hout)
__global__ void __launch_bounds__(256) selfloop_kernel(const float* __restrict__ hin,
                                                       float* __restrict__ hout,
                                                       const float* __restrict__ dinv) {
    long long t = (long long)blockIdx.x * 256 + threadIdx.x;
    int i = (int)(t >> 5);
    int q = (int)(t & 31);
    if (i >= NN) return;
    float s = dinv[i];
    s = s * s;
    const float4 v = *(const float4*)(hin + (size_t)i * C + q * 4);
    float4 r;
    r.x = v.x * s; r.y = v.y * s; r.z = v.z * s; r.w = v.w * s;
    *(float4*)(hout + (size_t)i * C + q * 4) = r;
}

// one wave per edge: lanes each carry 4 channels; gather 512B coalesced,
// scatter via native f32 atomics (L2-resident working set on MI455X's 192MB L2)
__global__ void __launch_bounds__(256) scatter_kernel(const long long* __restrict__ ei,
                                                      const float* __restrict__ nrm,
                                                      const float* __restrict__ hin,
                                                      float* __restrict__ hout) {
    long long t = (long long)blockIdx.x * 256 + threadIdx.x;
    int e = (int)(t >> 5);
    int q = (int)(t & 31);
    if (e >= NE) return;
    int s = (int)ei[e];                      // wave-uniform -> broadcast load
    int d = (int)ei[(size_t)NE + e];
    float w = nrm[e];
    const float4 v = *(const float4*)(hin + (size_t)s * C + q * 4);
    float* o = hout + (size_t)d * C + q * 4;
    atomAddF(o + 0, v.x * w);
    atomAddF(o + 1, v.y * w);
    atomAddF(o + 2, v.z * w);
    atomAddF(o + 3, v.w * w);
}

// out[m][n] = sum_k h[m][k] * W[n][k] + b[n], via V_WMMA_F32_16X16X4_F32
// one wave computes a 16x16 output tile; 6250 x 8 tiles, K stepped by 4 (32 WMMAs/tile)
__global__ void __launch_bounds__(256) gemm_wmma_kernel(const float* __restrict__ h,
                                                        const float* __restrict__ W,
                                                        const float* __restrict__ bias,
                                                        float* __restrict__ out) {
    int wave = (blockIdx.x * 256 + threadIdx.x) >> 5;
    int lane = threadIdx.x & 31;
    int tn = wave & 7;        // 8 tiles across 128 output channels
    int tm = wave >> 3;       // 6250 row tiles
    int half = lane >> 4;     // 0: lanes 0-15, 1: lanes 16-31
    int l15  = lane & 15;

    int arow = tm * 16 + l15;             // A: M index for this lane
    int bcol = tn * 16 + l15;             // B: N index for this lane

    // init accumulator with bias: VGPR r holds (M = r + half*8, N = bcol)
    float bv = bias[bcol];
    v8f c;
#pragma unroll
    for (int r = 0; r < 8; ++r) c[r] = bv;

    // lane-relative K offset per ISA A/B layout: lanes 16-31 carry K+2, K+3
    const float* aptr = h + (size_t)arow * C + half * 2;
    const float* bptr = W + (size_t)bcol * C + half * 2;   // B[k][n] = W[n*C + k]

#pragma unroll
    for (int k0 = 0; k0 < C; k0 += 4) {
        v2f a = *(const v2f*)(aptr + k0);
        v2f b = *(const v2f*)(bptr + k0);
        c = __builtin_amdgcn_wmma_f32_16x16x4_f32(
                /*neg_a=*/false, a, /*neg_b=*/false, b,
                /*c_mod=*/(short)0, c, /*reuse_a=*/false, /*reuse_b=*/false);
    }

    // store: VGPR r -> row (tm*16 + half*8 + r), col bcol
    float* obase = out + ((size_t)tm * 16 + half * 8) * C + tn * 16 + l15;
#pragma unroll
    for (int r = 0; r < 8; ++r) obase[(size_t)r * C] = c[r];
}

extern "C" void kernel_launch(void* const* d_in, const int* in_sizes, int n_in,
                              void* d_out, int out_size, void* d_ws, size_t ws_size,
                              hipStream_t stream) {
    const float*     x  = (const float*)d_in[0];       // [NN, C]
    const long long* ei = (const long long*)d_in[1];   // [2, NE] int64
    const float*     W  = (const float*)d_in[2];       // [C, C]
    const float*     b  = (const float*)d_in[3];       // [C]
    float*           out = (float*)d_out;              // [NN, C]

    // workspace layout (floats): hA | hB | dinv | norm  (~109 MB)
    float* hA   = (float*)d_ws;
    float* hB   = hA + (size_t)NN * C;
    float* dinv = hB + (size_t)NN * C;
    float* nrm  = dinv + NN;

    const int nb_nodes = (NN + 255) / 256;
    const int nb_edges = (NE + 255) / 256;
    const int nb_self  = (int)(((long long)NN * 32 + 255) / 256);
    const int nb_scat  = (int)(((long long)NE * 32 + 255) / 256);

    // 1) normalization
    deg_init_kernel<<<nb_nodes, 256, 0, stream>>>(dinv);              // dinv buffer doubles as deg
    deg_count_kernel<<<nb_edges, 256, 0, stream>>>(ei, dinv);
    dinv_kernel<<<nb_nodes, 256, 0, stream>>>(dinv);
    norm_kernel<<<nb_edges, 256, 0, stream>>>(ei, dinv, nrm);

    // 2) K_HOPS = 3 propagation, ping-pong x -> hA -> hB -> hA
    selfloop_kernel<<<nb_self, 256, 0, stream>>>(x, hA, dinv);
    scatter_kernel<<<nb_scat, 256, 0, stream>>>(ei, nrm, x, hA);

    selfloop_kernel<<<nb_self, 256, 0, stream>>>(hA, hB, dinv);
    scatter_kernel<<<nb_scat, 256, 0, stream>>>(ei, nrm, hA, hB);

    selfloop_kernel<<<nb_self, 256, 0, stream>>>(hB, hA, dinv);
    scatter_kernel<<<nb_scat, 256, 0, stream>>>(ei, nrm, hB, hA);

    // 3) final linear transform with f32 WMMA
    gemm_wmma_kernel<<<6250, 256, 0, stream>>>(hA, W, b, out);
}